// FeatureHasher_65687229825789
// MI455X (gfx1250) — compile-verified
//
#include <hip/hip_runtime.h>

typedef __attribute__((ext_vector_type(2))) float v2f;
typedef __attribute__((ext_vector_type(8))) float v8f;

#define NFEAT 1000000u
#define DM    64
#define NN    200
#define WAVES 8

// One block = 16 output rows (b0..b0+15). Eight wave32's stripe the n-loop.
// Each (b,n) gather event is one K-slice of V_WMMA_F32_16X16X4_F32:
//   A[m,k] = sign*val  iff m == owning b   (one-hot row)
//   B[k,d] = E[bucket(b_k,n), d0+d]
// so D accumulates sign*val * E[bucket] into exactly row b. Four d-tiles of 16
// cover D_MODEL=64; each 256B embedding row is read exactly once (64B/tile,
// 16 consecutive dwords per lane-half -> fully coalesced).
__global__ __launch_bounds__(256) void fh_wmma_kernel(
    const int* __restrict__ idx, const float* __restrict__ val,
    const float* __restrict__ emb, float* __restrict__ out)
{
    __shared__ float red[WAVES][16 * DM];   // 8 * 4KB = 32KB

    const int tid  = threadIdx.x;
    const int wave = tid >> 5;      // 0..7
    const int lane = tid & 31;
    const int m    = lane & 15;     // A-row / B-column index within tile
    const int hi   = lane >> 4;     // lane half selects K pair {2hi, 2hi+1}
    const int b0   = blockIdx.x * 16;

    v8f acc[4] = {};                // four 16x16 f32 d-tiles

    for (int n = wave; n < NN; n += WAVES) {
        // Per-lane A data: this lane's row m owns event (b0+m, n).
        const int   ii = idx[(size_t)(b0 + m) * NN + n];
        const float vv = val[(size_t)(b0 + m) * NN + n];
        const float sv = (ii & 1) ? vv : -vv;   // sign = 2*(idx%2)-1

        #pragma unroll
        for (int g = 0; g < 4; ++g) {           // K-groups: b = 4g..4g+3
            const int kb = 4 * g;
            // f32 16x4 A layout: VGPR0 = K=2hi, VGPR1 = K=2hi+1 (rows = lane&15)
            v2f a;
            a.x = (m == kb + 2 * hi)     ? sv : 0.0f;
            a.y = (m == kb + 2 * hi + 1) ? sv : 0.0f;

            // Rows this lane-half supplies for B (K = 2hi, 2hi+1)
            const unsigned i0 = (unsigned)idx[(size_t)(b0 + kb + 2 * hi)     * NN + n];
            const unsigned i1 = (unsigned)idx[(size_t)(b0 + kb + 2 * hi + 1) * NN + n];
            const float* r0 = emb + (size_t)(i0 % NFEAT) * DM;
            const float* r1 = emb + (size_t)(i1 % NFEAT) * DM;

            #pragma unroll
            for (int t = 0; t < 4; ++t) {       // d-tiles: columns 16t..16t+15
                v2f bm;
                bm.x = r0[t * 16 + m];          // 16 consecutive dwords/half
                bm.y = r1[t * 16 + m];
                acc[t] = __builtin_amdgcn_wmma_f32_16x16x4_f32(
                    /*neg_a=*/false, a, /*neg_b=*/false, bm,
                    /*c_mod=*/(short)0, acc[t],
                    /*reuse_a=*/false, /*reuse_b=*/false);
            }
        }
    }

    // Spill partial tiles: C/D layout -> VGPR v holds rows v (lanes 0-15) and
    // v+8 (lanes 16-31), column = lane&15.
    #pragma unroll
    for (int v = 0; v < 8; ++v) {
        const int row = v + 8 * hi;
        #pragma unroll
        for (int t = 0; t < 4; ++t)
            red[wave][row * DM + t * 16 + m] = acc[t][v];
    }
    __syncthreads();

    // Cross-wave reduction + global store of the 16x64 tile.
    for (int o = tid; o < 16 * DM; o += 256) {
        float s = 0.0f;
        #pragma unroll
        for (int w = 0; w < WAVES; ++w) s += red[w][o];
        out[(size_t)b0 * DM + o] = s;
    }
}

extern "C" void kernel_launch(void* const* d_in, const int* in_sizes, int n_in,
                              void* d_out, int out_size, void* d_ws, size_t ws_size,
                              hipStream_t stream) {
    const int*   idx = (const int*)  d_in[0];   // [4096*200] int32
    const float* val = (const float*)d_in[1];   // [4096*200] f32
    const float* emb = (const float*)d_in[2];   // [1e6*64]   f32
    float*       out = (float*)d_out;           // [4096*64]  f32
    (void)in_sizes; (void)n_in; (void)out_size; (void)d_ws; (void)ws_size;

    fh_wmma_kernel<<<4096 / 16, 256, 0, stream>>>(idx, val, emb, out);
}